// GatedStructuralEmbedder_72782515798137
// MI455X (gfx1250) — compile-verified
//
#include <hip/hip_runtime.h>
#include <hip/hip_bf16.h>

// Problem constants (match reference)
#define NNODES 16384
#define KSTR   32
#define DIM    64
#define DINP   66      // D + 2 (c, c_norm)
#define DPAD   68      // padded to multiple of 4 for 16x16x4 k-steps
#define HH     64
#define G3     192     // 3*H
#define GIP    260     // gi LDS row pitch in dwords: 64 h-slots * 4 + 4 pad
#define NB     4       // nodes per workgroup
#define ROWS   (NB * KSTR)   // 128 GEMM rows per workgroup
#define NAGG   3
#define KSTEPS (DPAD / 4)    // 17
#define NT     256           // threads per block (8 waves)

typedef float v8f __attribute__((ext_vector_type(8)));
typedef float v2f __attribute__((ext_vector_type(2)));

__device__ __forceinline__ float fsigmoid(float x) {
    return 1.0f / (1.0f + __expf(-x));
}
__device__ __forceinline__ float ftanh(float x) {
    // 1 - 2/(e^{2x}+1): stable for |x| large (e->inf => 1, e->0 => -1)
    float e = __expf(2.0f * x);
    return 1.0f - 2.0f / (e + 1.0f);
}

__global__ __launch_bounds__(NT, 1) void gse_fused_kernel(
    const int*   __restrict__ indices,  // [N,K]
    const int*   __restrict__ counts,   // [N,K]
    const float* __restrict__ matrix,   // [V,D]
    const float* __restrict__ W_ih,     // [3H, D+2]
    const float* __restrict__ b_ih,     // [3H]
    const float* __restrict__ W_hh,     // [3H, H]
    const float* __restrict__ b_hh,     // [3H]
    const float* __restrict__ W_out,    // [H, H]
    const float* __restrict__ b_out,    // [H]
    float*       __restrict__ out)      // [N, H]
{
    __shared__ float Ws[G3 * DPAD];     // W_ih padded (B-matrix source)    52,224 B
    __shared__ float xA[ROWS * DPAD];   // gathered+concat rows (A source)  34,816 B
    __shared__ float gi[ROWS * GIP];    // pre-activations [row][h*4+gate] 133,120 B
    __shared__ float bih[G3];
    __shared__ float hid[NB * HH];      // current hidden state
    __shared__ float gh[NB * G3];       // hidden-gate pre-activations
    __shared__ float cL[ROWS];          // log2(count+1)

    const int t = threadIdx.x;          // 0..255
    const int node0 = blockIdx.x * NB;

    // ---- Phase 0: stage W_ih (padded) + b_ih into LDS; compute c ----
    for (int i = t; i < G3 * DPAD; i += NT) {
        int g = i / DPAD, d = i - g * DPAD;
        Ws[i] = (d < DINP) ? W_ih[g * DINP + d] : 0.0f;
    }
    if (t < G3) bih[t] = b_ih[t];
    if (t < ROWS) {
        int n = t >> 5, k = t & 31;
        int cnt = counts[(node0 + n) * KSTR + k];
        cL[t] = __log2f((float)cnt + 1.0f);
    }
    hid[t] = 0.0f;                       // init hidden state (NB*HH == 256)
    __syncthreads();

    // ---- Phase 1: build A rows (embedding gather + c + c_norm) ----
    // two threads per row: half 0 -> cols 0..31, half 1 -> cols 32..63 + extras
    {
        int r = t >> 1, half = t & 1;    // r: 0..127
        int n = r >> 5, k = r & 31;
        int idx = indices[(node0 + n) * KSTR + k];
        const float4* src = reinterpret_cast<const float4*>(
            matrix + (size_t)idx * DIM + half * 32);
        float* dst = &xA[r * DPAD + half * 32];
        #pragma unroll
        for (int j = 0; j < 8; ++j) {    // 32 floats = 8 x float4
            float4 v = src[j];
            dst[4 * j + 0] = v.x; dst[4 * j + 1] = v.y;
            dst[4 * j + 2] = v.z; dst[4 * j + 3] = v.w;
        }
        if (half) {
            float cs = 0.0f;
            #pragma unroll
            for (int j = 0; j < KSTR; ++j) cs += cL[(n << 5) + j];
            float c = cL[r];
            float* row = &xA[r * DPAD];
            row[64] = c; row[65] = c / cs; row[66] = 0.0f; row[67] = 0.0f;
        }
    }
    __syncthreads();

    // ---- Phase 2: gi = xA @ W_ih^T + b_ih via V_WMMA_F32_16X16X4_F32 ----
    {
        const int w    = t >> 5;          // wave id 0..7 -> one M-tile each
        const int lane = t & 31;
        const int lrow = lane & 15;       // M (for A) / N (for B) within tile
        const int cA   = (lane >> 4) * 2; // lanes 0-15 -> k0,k1; 16-31 -> k2,k3
        const int mt   = w;               // 8 M-tiles, 8 waves

        v2f aF[KSTEPS];
        const int rA = mt * 16 + lrow;
        #pragma unroll
        for (int s = 0; s < KSTEPS; ++s)
            aF[s] = *reinterpret_cast<const v2f*>(&xA[rA * DPAD + s * 4 + cA]);

        for (int nt = 0; nt < 12; ++nt) {             // 192 / 16 N-tiles
            float b = bih[nt * 16 + lrow];
            v8f acc = { b, b, b, b, b, b, b, b };     // bias seed (per output col)
            #pragma unroll
            for (int s = 0; s < KSTEPS; ++s) {
                v2f bF = *reinterpret_cast<const v2f*>(
                    &Ws[(nt * 16 + lrow) * DPAD + s * 4 + cA]);
                acc = __builtin_amdgcn_wmma_f32_16x16x4_f32(
                    false, aF[s], false, bF, (short)0, acc, false, false);
            }
            // C/D layout: VGPR v, lanes0-15 -> M=v, lanes16-31 -> M=v+8; N = lane%16
            // Store into gate-interleaved layout: gi[row][ (col%64)*4 + col/64 ]
            const int rbase = mt * 16 + ((lane >> 4) << 3);
            const int col   = nt * 16 + lrow;
            const int coff  = ((col & 63) << 2) + (col >> 6);
            #pragma unroll
            for (int v = 0; v < 8; ++v)
                gi[(rbase + v) * GIP + coff] = acc[v];
        }
    }
    __syncthreads();

    // ---- Phase 3: 3 GRU aggregations (gi reused, gh recomputed) ----
    for (int it = 0; it < NAGG; ++it) {
        // gh = hid @ W_hh^T + b_hh   (NB*192 = 768 outputs, 3 per thread)
        #pragma unroll
        for (int i = 0; i < 3; ++i) {
            int idx = t + i * NT;
            int n = idx / G3, g = idx - n * G3;
            const float* wr = W_hh + g * HH;   // L2-resident
            const float* hr = &hid[n * HH];
            float s = b_hh[g];
            #pragma unroll
            for (int h = 0; h < HH; ++h) s += hr[h] * wr[h];
            gh[idx] = s;
        }
        __syncthreads();

        // gating + mean over K; each thread owns one fixed (n,h) slot
        {
            int n = t >> 6, h = t & 63;
            float hcur = hid[t];
            float ghr = gh[n * G3 + h];
            float ghz = gh[n * G3 + 64 + h];
            float ghn = gh[n * G3 + 128 + h];
            float acc = 0.0f;
            const float* base = &gi[(n * KSTR) * GIP + (h << 2)];
            for (int k = 0; k < KSTR; ++k) {
                // one b128 load: {gi_r, gi_z, gi_n, pad}
                float4 g4 = *reinterpret_cast<const float4*>(base + k * GIP);
                float r  = fsigmoid(g4.x + ghr);
                float z  = fsigmoid(g4.y + ghz);
                float nn = ftanh  (g4.z + r * ghn);
                acc += (1.0f - z) * nn + z * hcur;
            }
            hid[t] = acc * (1.0f / KSTR);   // only owner reads/writes this slot
        }
        __syncthreads();
    }

    // ---- Phase 4: out = hid @ W_out^T + b_out (one output per thread) ----
    {
        int n = t >> 6, o = t & 63;
        const float* wr = W_out + o * HH;
        const float* hr = &hid[n * HH];
        float s = b_out[o];
        #pragma unroll
        for (int h = 0; h < HH; ++h) s += hr[h] * wr[h];
        out[(size_t)(node0 + n) * HH + o] = s;
    }
}

extern "C" void kernel_launch(void* const* d_in, const int* in_sizes, int n_in,
                              void* d_out, int out_size, void* d_ws, size_t ws_size,
                              hipStream_t stream) {
    const int*   indices = (const int*)  d_in[0];
    const int*   counts  = (const int*)  d_in[1];
    const float* matrix  = (const float*)d_in[2];
    const float* W_ih    = (const float*)d_in[3];
    const float* b_ih    = (const float*)d_in[4];
    const float* W_hh    = (const float*)d_in[5];
    const float* b_hh    = (const float*)d_in[6];
    const float* W_out   = (const float*)d_in[7];
    const float* b_out   = (const float*)d_in[8];
    float* out = (float*)d_out;

    dim3 grid(NNODES / NB);   // 4096 workgroups of 4 nodes
    dim3 block(NT);           // 8 waves (wave32) -> 2 waves per SIMD32
    gse_fused_kernel<<<grid, block, 0, stream>>>(
        indices, counts, matrix, W_ih, b_ih, W_hh, b_hh, W_out, b_out, out);
}